// DynamicConvAttention3D_39788577030184
// MI455X (gfx1250) — compile-verified
//
#include <hip/hip_runtime.h>

typedef float v2f __attribute__((ext_vector_type(2)));
typedef float v8f __attribute__((ext_vector_type(8)));
typedef float f4  __attribute__((ext_vector_type(4)));

#define SPATIAL (64*64*64)        // 262144 voxels per (b, channel)
#define NSPLIT  16
#define SEG     (SPATIAL / NSPLIT) // 16384 floats per split

// ---------------------------------------------------------------------------
// Pass 1: partial sums for global average pool of cat([g, x]).
// channel = b*128 + c  (c<64 -> g channel c ; c>=64 -> x channel c-64)
// grid.x = 256 channels * 16 splits = 4096 blocks, 256 threads.
// x is streamed with non-temporal loads to bias L2 residency toward g
// (g is re-read in pass 3; 134 MB fits in the 192 MB L2).
// ---------------------------------------------------------------------------
__global__ __launch_bounds__(256) void k_pool(const float* __restrict__ g,
                                              const float* __restrict__ x,
                                              float* __restrict__ partial) {
    __shared__ float red[256];
    int channel = blockIdx.x >> 4;
    int split   = blockIdx.x & 15;
    int b = channel >> 7;
    int c = channel & 127;
    bool isg = (c < 64);
    size_t chan = (size_t)b * 64 + (size_t)(isg ? c : (c - 64));
    const float* src = (isg ? g : x) + chan * (size_t)SPATIAL;
    const f4* src4 = (const f4*)src + (size_t)split * (SEG / 4);

    int t = threadIdx.x;
    float sum = 0.f;
#pragma unroll
    for (int i = 0; i < SEG / 4 / 256; ++i) {   // 16 float4 per thread
        f4 v;
        if (isg) v = src4[i * 256 + t];
        else     v = __builtin_nontemporal_load(&src4[i * 256 + t]);
        sum += v.x + v.y + v.z + v.w;
    }
    red[t] = sum;
    __syncthreads();
    for (int s = 128; s > 0; s >>= 1) {
        if (t < s) red[t] += red[t + s];
        __syncthreads();
    }
    if (t == 0) partial[channel * NSPLIT + split] = red[0];
}

// ---------------------------------------------------------------------------
// Pass 2 (single block, 8 waves): finish pooled means, run the gating MLP.
// The h[2,32] @ w2^T[32,4096] GEMM uses V_WMMA_F32_16X16X4_F32 (M padded to
// 16, K-loop of 8 covers F_INT=32, 256 N-tiles over the 4096 outputs).
// Then dw = sigmoid(.) is folded immediately into v[b,g] = sum_l dw*psi_w[l]
// via LDS float atomics — `attended` is never materialized.
// ---------------------------------------------------------------------------
__global__ __launch_bounds__(256) void k_mlp(const float* __restrict__ partial,
                                             const float* __restrict__ w1,
                                             const float* __restrict__ b1,
                                             const float* __restrict__ w2,
                                             const float* __restrict__ b2,
                                             const float* __restrict__ psi_w,
                                             float* __restrict__ vout) {
    __shared__ float pooled[256];     // [b*128 + c]
    __shared__ float hbuf[16 * 32];   // A matrix, M=16 (rows 2..15 zero) x K=32
    __shared__ float vsh[128];        // v[b*64 + g]
    int t = threadIdx.x;

    {   // mean over spatial: sum the 16 partials per channel
        float s = 0.f;
#pragma unroll
        for (int i = 0; i < NSPLIT; ++i) s += partial[t * NSPLIT + i];
        pooled[t] = s * (1.0f / (float)SPATIAL);
    }
    if (t < 128) vsh[t] = 0.f;
    hbuf[t] = 0.f; hbuf[t + 256] = 0.f;   // zero 512-entry A buffer
    __syncthreads();

    if (t < 64) {  // h[m][k] = relu(pooled[m] . w1[k] + b1[k]);  m=0..1, k=0..31
        int m = t >> 5;
        int k = t & 31;
        const float* prow = &pooled[m * 128];
        const float* wrow = &w1[k * 128];
        float acc = b1[k];
        for (int c = 0; c < 128; ++c) acc += prow[c] * wrow[c];
        hbuf[m * 32 + k] = fmaxf(acc, 0.f);
    }
    __syncthreads();

    int lane = t & 31;
    int wave = t >> 5;
    int row  = lane & 15;
    int hi   = lane >> 4;     // 0: K+{0,1}, 1: K+{2,3}  (ISA 32-bit A layout)

    // A fragments for the 8 K-steps (identical for every tile)
    v2f afrag[8];
#pragma unroll
    for (int kk = 0; kk < 8; ++kk) {
        int k0 = kk * 4 + hi * 2;
        afrag[kk].x = hbuf[row * 32 + k0];
        afrag[kk].y = hbuf[row * 32 + k0 + 1];
    }

    // 4096 outputs -> 256 tiles of N=16; 8 waves -> 32 tiles per wave
    for (int tt = 0; tt < 32; ++tt) {
        int tile    = wave + (tt << 3);
        int colbase = tile * 16;
        v8f c = {};
#pragma unroll
        for (int kk = 0; kk < 8; ++kk) {
            int k0 = kk * 4 + hi * 2;
            v2f bfrag;            // B[k][n] = w2[(colbase+n)*32 + k]
            bfrag.x = w2[(size_t)(colbase + row) * 32 + k0];
            bfrag.y = w2[(size_t)(colbase + row) * 32 + k0 + 1];
            c = __builtin_amdgcn_wmma_f32_16x16x4_f32(
                    false, afrag[kk], false, bfrag, (short)0, c, false, false);
        }
        if (lane < 16) {          // rows M=0,1 live in c[0], c[1] for lanes 0-15
            int j  = colbase + lane;
            int l  = j & 63;
            int gg = j >> 6;
            float pw = psi_w[l];
            float d0 = c[0] + b2[j];
            float d1 = c[1] + b2[j];
            d0 = 1.0f / (1.0f + __expf(-d0));
            d1 = 1.0f / (1.0f + __expf(-d1));
            atomicAdd(&vsh[gg], d0 * pw);        // b = 0
            atomicAdd(&vsh[64 + gg], d1 * pw);   // b = 1
        }
    }
    __syncthreads();
    if (t < 128) vout[t] = vsh[t];
}

// ---------------------------------------------------------------------------
// Pass 3: per voxel sval = sum_g g[b,g,s]*v[b,g]; scale = sigmoid(BN(sval));
// out[b,l,s] = x[b,l,s]*scale.  float4 per thread, v broadcast from LDS,
// x/out streamed non-temporally (never re-read), g reads regular (L2 hits
// expected from pass 1).  grid = 512 blocks (256 per batch) x 256 threads.
// ---------------------------------------------------------------------------
__global__ __launch_bounds__(256) void k_apply(const float* __restrict__ g,
                                               const float* __restrict__ x,
                                               const float* __restrict__ vbuf,
                                               const float* __restrict__ psi_b,
                                               const float* __restrict__ bn_gamma,
                                               const float* __restrict__ bn_beta,
                                               const float* __restrict__ bn_mean,
                                               const float* __restrict__ bn_var,
                                               float* __restrict__ out) {
    __shared__ float vsh[64];
    int t  = threadIdx.x;
    int b  = blockIdx.x >> 8;                       // 256 blocks per batch
    int sv = ((blockIdx.x & 255) << 8) + t;         // float4 index 0..65535
    if (t < 64) vsh[t] = vbuf[b * 64 + t];
    __syncthreads();

    float a_bn  = bn_gamma[0] * rsqrtf(bn_var[0] + 1e-5f);
    float shift = (psi_b[0] - bn_mean[0]) * a_bn + bn_beta[0];

    const f4* gb = (const f4*)(g + (size_t)b * 64 * SPATIAL);
    const f4* xb = (const f4*)(x + (size_t)b * 64 * SPATIAL);
    f4*       ob = (f4*)out + (size_t)b * 64 * (SPATIAL / 4);

    f4 s = {0.f, 0.f, 0.f, 0.f};
#pragma unroll 8
    for (int gc = 0; gc < 64; ++gc) {
        f4 gv = gb[(size_t)gc * (SPATIAL / 4) + sv];
        s += gv * vsh[gc];
    }
    f4 z = s * a_bn + shift;
    f4 sc;
    sc.x = 1.f / (1.f + __expf(-z.x));
    sc.y = 1.f / (1.f + __expf(-z.y));
    sc.z = 1.f / (1.f + __expf(-z.z));
    sc.w = 1.f / (1.f + __expf(-z.w));

    for (int l = 0; l < 64; ++l) {
        f4 xv = __builtin_nontemporal_load(&xb[(size_t)l * (SPATIAL / 4) + sv]);
        __builtin_nontemporal_store(xv * sc, &ob[(size_t)l * (SPATIAL / 4) + sv]);
    }
}

extern "C" void kernel_launch(void* const* d_in, const int* in_sizes, int n_in,
                              void* d_out, int out_size, void* d_ws, size_t ws_size,
                              hipStream_t stream) {
    (void)in_sizes; (void)n_in; (void)out_size; (void)ws_size;
    const float* g        = (const float*)d_in[0];
    const float* x        = (const float*)d_in[1];
    const float* w1       = (const float*)d_in[2];
    const float* b1       = (const float*)d_in[3];
    const float* w2       = (const float*)d_in[4];
    const float* b2       = (const float*)d_in[5];
    const float* psi_w    = (const float*)d_in[6];
    const float* psi_b    = (const float*)d_in[7];
    const float* bn_gamma = (const float*)d_in[8];
    const float* bn_beta  = (const float*)d_in[9];
    const float* bn_mean  = (const float*)d_in[10];
    const float* bn_var   = (const float*)d_in[11];
    float* out = (float*)d_out;

    float* partial = (float*)d_ws;        // 256*16 = 4096 floats (fully written)
    float* vbuf    = partial + 4096;      // 128 floats

    hipLaunchKernelGGL(k_pool,  dim3(4096), dim3(256), 0, stream, g, x, partial);
    hipLaunchKernelGGL(k_mlp,   dim3(1),    dim3(256), 0, stream,
                       partial, w1, b1, w2, b2, psi_w, vbuf);
    hipLaunchKernelGGL(k_apply, dim3(512),  dim3(256), 0, stream,
                       g, x, vbuf, psi_b, bn_gamma, bn_beta, bn_mean, bn_var, out);
}